// FeatureSampler_83021717832240
// MI455X (gfx1250) — compile-verified
//
#include <hip/hip_runtime.h>
#include <math.h>

// ---------------------------------------------------------------------------
// FeatureSampler for MI455X (gfx1250).
// Pass 1: project BEV reference points -> per (level,view,pixel) image coords
//         + validity mask (written straight into d_out's mask slab).
// Pass 2: one block per (level,view,channel). Stage the 2D feature plane
//         (<= 30 KB) into LDS via gfx1250 async-to-LDS (ASYNCcnt), then do
//         the divergent bilinear gathers from LDS (ds_load, bank-parallel)
//         with coalesced float4 stores to the 403 MB output.
// ---------------------------------------------------------------------------

#define NVIEW   6
#define NLEV    4
#define NCH     256
#define BEV_H   128
#define BEV_W   128
#define NP      (BEV_H * BEV_W)          // 16384 pixels
#define NG      (NP / 4)                 // 4096 float4 groups per (l,v,c)
#define MAXPLANE 7680                    // 64*120 floats = 30 KB (level 0)

__device__ __constant__ int kHf[NLEV] = {64, 32, 16, 8};
__device__ __constant__ int kWf[NLEV] = {120, 60, 30, 15};

#if defined(__has_builtin)
#if __has_builtin(__builtin_amdgcn_global_load_async_to_lds_b128)
#define HAVE_ASYNC_LDS 1
#endif
#endif
#ifndef HAVE_ASYNC_LDS
#define HAVE_ASYNC_LDS 0
#endif

#if HAVE_ASYNC_LDS
typedef int v4i_t __attribute__((ext_vector_type(4)));
typedef __attribute__((address_space(1))) v4i_t* g_v4i_ptr;   // global src
typedef __attribute__((address_space(3))) v4i_t* l_v4i_ptr;   // LDS dst
#endif

// ---------------------------------------------------------------------------
// Pass 1: coordinates + mask.  One thread per (level, view, pixel).
// ---------------------------------------------------------------------------
__global__ void fs_coords_kernel(const float* __restrict__ refpts,     // (1,3,128,128)
                                 const float* __restrict__ homography, // (6,4,4)
                                 const float* __restrict__ bev_range,  // (6,)
                                 const int*   __restrict__ orig_h_p,
                                 const int*   __restrict__ orig_w_p,
                                 float*       __restrict__ coords,     // [4][6][NP][2]
                                 float*       __restrict__ mask_out)   // [6][4][NP]
{
    int idx = blockIdx.x * blockDim.x + threadIdx.x;
    const int total = NLEV * NVIEW * NP;
    if (idx >= total) return;

    int l   = idx / (NVIEW * NP);
    int rem = idx - l * (NVIEW * NP);
    int v   = rem / NP;
    int p   = rem - v * NP;

    const int   Hf = kHf[l];
    const int   Wf = kWf[l];
    const float sh = (float)Hf / (float)(*orig_h_p);   // exact powers of two
    const float sw = (float)Wf / (float)(*orig_w_p);

    const float lo0 = bev_range[0], lo1 = bev_range[1], lo2 = bev_range[2];
    const float hi0 = bev_range[3], hi1 = bev_range[4], hi2 = bev_range[5];

    // sigmoid(reference_points) then scale to bev range
    const float r0 = refpts[0 * NP + p];
    const float r1 = refpts[1 * NP + p];
    const float r2 = refpts[2 * NP + p];
    const float s0 = 1.0f / (1.0f + expf(-r0));
    const float s1 = 1.0f / (1.0f + expf(-r1));
    const float s2 = 1.0f / (1.0f + expf(-r2));
    const float x  = s0 * (hi0 - lo0) + lo0;
    const float y  = s1 * (hi1 - lo1) + lo1;
    const float z  = s2 * (hi2 - lo2) + lo2;

    // homo = view(sw,sh) @ homography[v]; rows 0/1 scaled, row 2 unchanged.
    const float* Hm = homography + v * 16;
    const float c0 = (sw * Hm[0]) * x + (sw * Hm[1]) * y + (sw * Hm[2])  * z + (sw * Hm[3]);
    const float c1 = (sh * Hm[4]) * x + (sh * Hm[5]) * y + (sh * Hm[6])  * z + (sh * Hm[7]);
    const float c2 =        Hm[8] * x +        Hm[9] * y +        Hm[10] * z +        Hm[11];

    const float zc = fmaxf(c2, 0.05f);
    const float X  = c0 / zc;
    const float Y  = c1 / zc;

    // coords layout: [(l*6+v)][p] as float2
    float2* cp = (float2*)coords;
    cp[(size_t)(l * NVIEW + v) * NP + p] = make_float2(X, Y);

    // mask layout: (6 views, 4 levels, 128, 128)
    const bool m = (X >= 0.0f) && (X < (float)Wf) && (Y >= 0.0f) && (Y < (float)Hf);
    mask_out[(size_t)v * (NLEV * NP) + (size_t)l * NP + p] = m ? 1.0f : 0.0f;
}

// ---------------------------------------------------------------------------
// One bilinear tap out of the LDS-staged plane: clamp (always in-bounds),
// zero if outside.  Lowered to ds_load_b32 after inlining.
// ---------------------------------------------------------------------------
__device__ __forceinline__ float fs_tap(const float* __restrict__ splane,
                                        float xf, float yf, int Hf, int Wf)
{
    const bool valid = (xf >= 0.0f) && (xf <= (float)(Wf - 1)) &&
                       (yf >= 0.0f) && (yf <= (float)(Hf - 1));
    const float xc = fminf(fmaxf(xf, 0.0f), (float)(Wf - 1));
    const float yc = fminf(fmaxf(yf, 0.0f), (float)(Hf - 1));
    const int   xi = (int)xc;
    const int   yi = (int)yc;
    const float val = splane[yi * Wf + xi];
    return valid ? val : 0.0f;
}

// ---------------------------------------------------------------------------
// Pass 2: one block (256 threads) per (l, v, c).
//   Stage plane -> LDS (async), then 16 iterations x 4 pixels per thread.
// ---------------------------------------------------------------------------
__global__ void __launch_bounds__(256)
fs_sample_kernel(const float* __restrict__ f0,
                 const float* __restrict__ f1,
                 const float* __restrict__ f2,
                 const float* __restrict__ f3,
                 const float* __restrict__ coords,
                 float*       __restrict__ out)
{
    __shared__ __align__(16) float splane[MAXPLANE];

    const int t  = blockIdx.x;          // (l*6+v)*256 + c
    const int c  = t & (NCH - 1);
    const int lv = t >> 8;              // l*6 + v
    const int v  = lv % NVIEW;
    const int l  = lv / NVIEW;

    const int Hf    = kHf[l];
    const int Wf    = kWf[l];
    const int plane = Hf * Wf;          // multiple of 4 for every level

    const float* feat  = (l == 0) ? f0 : (l == 1) ? f1 : (l == 2) ? f2 : f3;
    const float* gbase = feat + ((size_t)v * NCH + (size_t)c) * (size_t)plane;

    // ---- stage the whole (v,c) plane into LDS -----------------------------
    const int nvec = plane >> 2;        // float4 chunks
#if HAVE_ASYNC_LDS
    for (int i = threadIdx.x; i < nvec; i += 256) {
        __builtin_amdgcn_global_load_async_to_lds_b128(
            (g_v4i_ptr)(gbase + 4 * i),
            (l_v4i_ptr)(splane + 4 * i),
            0, 0);
    }
#if __has_builtin(__builtin_amdgcn_s_wait_asynccnt)
    __builtin_amdgcn_s_wait_asynccnt(0);
#else
    asm volatile("s_wait_asynccnt 0x0" ::: "memory");
#endif
    __syncthreads();
#else
    for (int i = threadIdx.x; i < nvec; i += 256) {
        ((float4*)splane)[i] = ((const float4*)gbase)[i];
    }
    __syncthreads();
#endif

    // ---- bilinear gather out of LDS ---------------------------------------
    const float2* cp = (const float2*)coords + (size_t)lv * NP;
    const size_t  dst_base =
        ((size_t)v * (NLEV * NCH) + (size_t)l * NCH + (size_t)c) * NP;

#pragma unroll 2
    for (int it = 0; it < 16; ++it) {
        const int g = it * 256 + (int)threadIdx.x;   // coalesced float4 group

        float acc[4];
#pragma unroll
        for (int i = 0; i < 4; ++i) {
            const float2 XY = cp[g * 4 + i];
            const float X  = XY.x;
            const float Y  = XY.y;
            const float x0 = floorf(X);
            const float y0 = floorf(Y);
            const float wx = X - x0;
            const float wy = Y - y0;

            const float v00 = fs_tap(splane, x0,        y0,        Hf, Wf);
            const float v10 = fs_tap(splane, x0 + 1.0f, y0,        Hf, Wf);
            const float v01 = fs_tap(splane, x0,        y0 + 1.0f, Hf, Wf);
            const float v11 = fs_tap(splane, x0 + 1.0f, y0 + 1.0f, Hf, Wf);

            acc[i] = v00 * (1.0f - wx) * (1.0f - wy)
                   + v10 * wx          * (1.0f - wy)
                   + v01 * (1.0f - wx) * wy
                   + v11 * wx          * wy;
        }

        *(float4*)(out + dst_base + (size_t)g * 4) =
            make_float4(acc[0], acc[1], acc[2], acc[3]);
    }
}

// ---------------------------------------------------------------------------
// Launch
// ---------------------------------------------------------------------------
extern "C" void kernel_launch(void* const* d_in, const int* in_sizes, int n_in,
                              void* d_out, int out_size, void* d_ws, size_t ws_size,
                              hipStream_t stream)
{
    const float* f0   = (const float*)d_in[0];
    const float* f1   = (const float*)d_in[1];
    const float* f2   = (const float*)d_in[2];
    const float* f3   = (const float*)d_in[3];
    const float* rp   = (const float*)d_in[4];
    const float* homo = (const float*)d_in[5];
    const float* bev  = (const float*)d_in[6];
    const int*   oh   = (const int*)d_in[7];
    const int*   ow   = (const int*)d_in[8];

    float* out      = (float*)d_out;
    float* coords   = (float*)d_ws;                                  // 3 MB scratch
    float* mask_out = out + (size_t)NVIEW * NLEV * NCH * NP;         // mask slab

    const int n1 = NLEV * NVIEW * NP;                                // 393216
    fs_coords_kernel<<<(n1 + 255) / 256, 256, 0, stream>>>(
        rp, homo, bev, oh, ow, coords, mask_out);

    const int nblk = NLEV * NVIEW * NCH;                             // 6144 blocks
    fs_sample_kernel<<<nblk, 256, 0, stream>>>(f0, f1, f2, f3, coords, out);
}